// Transformer_F_59871844106464
// MI455X (gfx1250) — compile-verified
//
#include <hip/hip_runtime.h>

// MI455X / gfx1250, wave32, WMMA f32 16x16x4.
// Requires ws_size >= 8*144*144*4 = 663,552 bytes (transposed padded weights).

typedef __attribute__((ext_vector_type(2))) float v2f;
typedef __attribute__((ext_vector_type(8))) float v8f;

#define NR 257        // N+1 rows of Z
#define NC 129        // D+1 cols of Z
#define RPAD 272      // 17 * 16
#define CPAD 144      // 9 * 16
#define SZS 145       // LDS stride in floats (conflict-free transposed reads)
#define NTHREADS 512
#define NWAVES 16
#define T1 81         // 9x9 tiles for 144x144 outputs
#define T4 153        // 17x9 tiles for 272x144 output
#define MAXT4 10      // ceil(153/16)
#define WMAT (144*144)

__device__ __forceinline__ v8f wmma_f32(v2f a, v2f b, v8f c) {
  // D = A(16x4,f32) * B(4x16,f32) + C(16x16,f32)
  return __builtin_amdgcn_wmma_f32_16x16x4_f32(false, a, false, b, (short)0, c,
                                               false, false);
}

// Build padded, transposed weights once per call:
//   wt[layer*2+0] = P_full^T (144x144 row-major, corner [128][128] = 1)
//   wt[layer*2+1] = Q_full^T (144x144 row-major, corner = 0)
__global__ void prep_weights(const float* __restrict__ allparam,
                             float* __restrict__ wt) {
  const int mi = blockIdx.x;            // 0..7 = layer*2 + which
  const int layer = mi >> 1, which = mi & 1;
  const float* src = allparam + ((size_t)layer * 2 + which) * 128 * 128;
  const float corner = (which == 0) ? 1.0f : 0.0f;
  float* dst = wt + (size_t)mi * WMAT;
  for (int s = threadIdx.x; s < WMAT; s += blockDim.x) {
    const int k = s / CPAD, m = s - k * CPAD;   // dst[k][m] = W_full[m][k]
    float v = 0.0f;
    if (k < 128 && m < 128) v = src[m * 128 + k];
    else if (k == 128 && m == 128) v = corner;
    dst[s] = v;
  }
}

__global__ __launch_bounds__(NTHREADS) void tf_layers_kernel(
    const float* __restrict__ Zin, const float* __restrict__ gamma,
    const float* __restrict__ wt, float* __restrict__ out) {
  extern __shared__ float lds[];
  float* Zl = lds;                       // RPAD x SZS  (157,760 B)
  float* Ub = lds + RPAD * SZS;          // CPAD x SZS  ( 83,520 B)
  float* Vb = Ub + CPAD * SZS;           // CPAD x SZS  ( 83,520 B)

  const int b = blockIdx.x;
  const int tid = threadIdx.x;
  const int w = tid >> 5;                // wave id 0..15 (wave32)
  const int lane = tid & 31;
  const int lm = lane & 15;              // row/col within tile
  const int lh = lane >> 4;              // k-half select (A/B), row+8 (C/D)
  const int kh = 2 * lh;                 // per-lane k offset

  const float* Zb = Zin + (size_t)b * NR * NC;
  float* outb = out + (size_t)b * NR * NC;

  // Zero LDS pads, then load Z (257x129) into padded 272x145 buffer.
  for (int s = tid; s < RPAD * SZS; s += NTHREADS) Zl[s] = 0.0f;
  __syncthreads();
  for (int s = tid; s < NR * NC; s += NTHREADS) {
    const int m = s / NC, n2 = s - m * NC;
    Zl[m * SZS + n2] = Zb[s];
  }
  __syncthreads();

  for (int layer = 0; layer < 4; ++layer) {
    const float* Pt = wt + (size_t)(layer * 2 + 0) * WMAT;   // P_full^T
    const float* Qt = wt + (size_t)(layer * 2 + 1) * WMAT;   // Q_full^T
    const float* gl = gamma + (size_t)layer * NR * NC;

    // ---- Step 1: U = G = Z^T * Z''  (k < 256 implements key-row mask) ----
    // Software-pipelined: issue next K-step's LDS loads before consuming
    // current operands so s_wait_dscnt overlaps a full iteration.
    for (int t = w; t < T1; t += NWAVES) {
      const int ti = t / 9, tj = t - (t / 9) * 9;
      const int i0 = ti * 16 + lm, j0 = tj * 16 + lm;
      v8f c = {0.f, 0.f, 0.f, 0.f, 0.f, 0.f, 0.f, 0.f};
      v2f a, bb;
      a[0]  = Zl[kh * SZS + i0];       a[1]  = Zl[(kh + 1) * SZS + i0];
      bb[0] = Zl[kh * SZS + j0];       bb[1] = Zl[(kh + 1) * SZS + j0];
      for (int k0 = 0; k0 < 252; k0 += 4) {
        const int k = k0 + 4 + kh;
        v2f an, bn;
        an[0] = Zl[k * SZS + i0];      an[1] = Zl[(k + 1) * SZS + i0];
        bn[0] = Zl[k * SZS + j0];      bn[1] = Zl[(k + 1) * SZS + j0];
        c = wmma_f32(a, bb, c);
        a = an; bb = bn;
      }
      c = wmma_f32(a, bb, c);
#pragma unroll
      for (int r = 0; r < 8; ++r)
        Ub[(ti * 16 + r + 8 * lh) * SZS + tj * 16 + lm] = c[r];
    }
    __syncthreads();

    // ---- Step 2: V = M1t = P_full * G   (K active = 129 -> 33 steps) ----
    for (int t = w; t < T1; t += NWAVES) {
      const int ti = t / 9, tj = t - (t / 9) * 9;
      const int i0 = ti * 16 + lm, j0 = tj * 16 + lm;
      v8f c = {0.f, 0.f, 0.f, 0.f, 0.f, 0.f, 0.f, 0.f};
      v2f a, bb;   // A[m][k] = P_full[m][k] = Pt[k][m] (coalesced global)
      a[0]  = Pt[kh * CPAD + i0];      a[1]  = Pt[(kh + 1) * CPAD + i0];
      bb[0] = Ub[kh * SZS + j0];       bb[1] = Ub[(kh + 1) * SZS + j0];
      for (int k0 = 0; k0 < 128; k0 += 4) {
        const int k = k0 + 4 + kh;
        v2f an, bn;
        an[0] = Pt[k * CPAD + i0];     an[1] = Pt[(k + 1) * CPAD + i0];
        bn[0] = Ub[k * SZS + j0];      bn[1] = Ub[(k + 1) * SZS + j0];
        c = wmma_f32(a, bb, c);
        a = an; bb = bn;
      }
      c = wmma_f32(a, bb, c);
#pragma unroll
      for (int r = 0; r < 8; ++r)
        Vb[(ti * 16 + r + 8 * lh) * SZS + tj * 16 + lm] = c[r];
    }
    __syncthreads();

    // ---- Step 3: U = M2t = M1t * Q_full^T ----
    for (int t = w; t < T1; t += NWAVES) {
      const int ti = t / 9, tj = t - (t / 9) * 9;
      const int i0 = ti * 16 + lm, j0 = tj * 16 + lm;
      v8f c = {0.f, 0.f, 0.f, 0.f, 0.f, 0.f, 0.f, 0.f};
      v2f a, bb;   // A = M1t (stride-145 LDS: conflict-free), B = Qt (global)
      a[0]  = Vb[i0 * SZS + kh];       a[1]  = Vb[i0 * SZS + kh + 1];
      bb[0] = Qt[kh * CPAD + j0];      bb[1] = Qt[(kh + 1) * CPAD + j0];
      for (int k0 = 0; k0 < 128; k0 += 4) {
        const int k = k0 + 4 + kh;
        v2f an, bn;
        an[0] = Vb[i0 * SZS + k];      an[1] = Vb[i0 * SZS + k + 1];
        bn[0] = Qt[k * CPAD + j0];     bn[1] = Qt[(k + 1) * CPAD + j0];
        c = wmma_f32(a, bb, c);
        a = an; bb = bn;
      }
      c = wmma_f32(a, bb, c);
#pragma unroll
      for (int r = 0; r < 8; ++r)
        Ub[(ti * 16 + r + 8 * lh) * SZS + tj * 16 + lm] = c[r];
    }
    __syncthreads();

    // ---- Step 4a: R = (Z * M2t^T) / 256 into registers (no Z write yet) ----
    v8f rc[MAXT4];
#pragma unroll
    for (int it = 0; it < MAXT4; ++it) {
      const int t = w + it * NWAVES;
      if (t < T4) {
        const int ti = t / 9, tj = t - (t / 9) * 9;
        const int i0 = ti * 16 + lm, j0 = tj * 16 + lm;
        v8f c = {0.f, 0.f, 0.f, 0.f, 0.f, 0.f, 0.f, 0.f};
        v2f a, bb;  // A = Z (strided, conflict-free), B = M2t^T (transposed)
        a[0]  = Zl[i0 * SZS + kh];     a[1]  = Zl[i0 * SZS + kh + 1];
        bb[0] = Ub[j0 * SZS + kh];     bb[1] = Ub[j0 * SZS + kh + 1];
        for (int k0 = 0; k0 < 128; k0 += 4) {
          const int k = k0 + 4 + kh;
          v2f an, bn;
          an[0] = Zl[i0 * SZS + k];    an[1] = Zl[i0 * SZS + k + 1];
          bn[0] = Ub[j0 * SZS + k];    bn[1] = Ub[j0 * SZS + k + 1];
          c = wmma_f32(a, bb, c);
          a = an; bb = bn;
        }
        c = wmma_f32(a, bb, c);
        rc[it] = c * (1.0f / 256.0f);
      }
    }
    __syncthreads();

    // ---- Step 4b: Z += R + acc ; acc += R*gamma (acc lives in d_out) ----
#pragma unroll
    for (int it = 0; it < MAXT4; ++it) {
      const int t = w + it * NWAVES;
      if (t < T4) {
        const int ti = t / 9, tj = t - (t / 9) * 9;
#pragma unroll
        for (int r = 0; r < 8; ++r) {
          const int m = ti * 16 + r + 8 * lh;
          const int n2 = tj * 16 + lm;
          if (m < NR && n2 < NC) {
            const float rv = rc[it][r];
            const int g = m * NC + n2;
            const float zold = Zl[m * SZS + n2];
            const float accp = (layer == 0) ? 0.0f : outb[g];
            const float znew = zold + rv + accp;
            Zl[m * SZS + n2] = znew;
            outb[g] = (layer < 3) ? (accp + rv * gl[g]) : znew;
          }
        }
      }
    }
    __syncthreads();
  }
}

extern "C" void kernel_launch(void* const* d_in, const int* in_sizes, int n_in,
                              void* d_out, int out_size, void* d_ws,
                              size_t ws_size, hipStream_t stream) {
  (void)in_sizes; (void)n_in; (void)out_size; (void)ws_size;
  const float* Z        = (const float*)d_in[0];
  const float* allparam = (const float*)d_in[1];
  const float* gamma    = (const float*)d_in[2];
  float* out = (float*)d_out;
  float* wt  = (float*)d_ws;   // 8 * 144*144 floats = 663,552 bytes

  prep_weights<<<8, 256, 0, stream>>>(allparam, wt);

  const size_t shmem = (size_t)(RPAD * SZS + 2 * CPAD * SZS) * sizeof(float);
  (void)hipFuncSetAttribute((const void*)tf_layers_kernel,
                            hipFuncAttributeMaxDynamicSharedMemorySize,
                            (int)shmem);
  tf_layers_kernel<<<512, NTHREADS, shmem, stream>>>(Z, gamma, wt, out);
}